// EnhanceImageProducer_79293686219249
// MI455X (gfx1250) — compile-verified
//
#include <hip/hip_runtime.h>
#include <math.h>

// CDNA5 / gfx1250: wave32, WMMA 16x16 tiles.
typedef __attribute__((ext_vector_type(2))) float v2f;
typedef __attribute__((ext_vector_type(8))) float v8f;

#define IMG_H  256
#define IMG_W  256
#define ORI_H  32
#define ORI_W  32
#define NANG   180
#define KS     25
#define KK     625                 // 25*25
#define HALO   12
#define TILE_W 16                  // pixels per wave (one M=16 GEMM strip)
#define TILE_H 8                   // one wave per pixel row, 8 waves per block
#define LW     (TILE_W + 2*HALO)   // 40
#define LH     (TILE_H + 2*HALO)   // 32

// One 25-wide kernel row of the K-reduction: 6 clean K=4 WMMAs + 1 tail WMMA
// (only K=24 real, carried in the .x slot of lanes 0-15, zero elsewhere).
// All loads are unconditional with immediate offsets; tail masking is a
// register select (v_cndmask), so EXEC stays all-1s around every WMMA.
__device__ __forceinline__
v8f gabor_row(const float* __restrict__ ta0,   // LDS:    &tile[(wv+ky)*LW + m]
              const float* __restrict__ rb0,   // global: &kf[ky*25]
              int kcol, bool lo, v8f acc)
{
    const float* ta = ta0 + kcol;              // +0 (lanes 0-15) or +2 (lanes 16-31)
    const float* rb = rb0 + kcol;
    #pragma unroll
    for (int kx = 0; kx < 24; kx += 4) {
        v2f a, b;
        a.x = ta[kx];                          // ds_load_b32, imm offset
        a.y = ta[kx + 1];
        b.x = rb[kx];                          // global_load_b32, imm offset
        b.y = rb[kx + 1];
        acc = __builtin_amdgcn_wmma_f32_16x16x4_f32(false, a, false, b,
                                                    (short)0, acc, false, false);
    }
    float av = ta0[24];                        // always in-bounds for all lanes
    float bv = rb0[24];
    v2f a, b;
    a.x = lo ? av : 0.f;  a.y = 0.f;
    b.x = lo ? bv : 0.f;  b.y = 0.f;
    return __builtin_amdgcn_wmma_f32_16x16x4_f32(false, a, false, b,
                                                 (short)0, acc, false, false);
}

// Routed Gabor conv:
//   idx[p] = orientation from bilinear-upsampled oris + atan2   (per pixel)
//   out[p] = <25x25 patch(p), gabor_kernel[idx[p]]>             (per pixel)
// Per 16-pixel strip: D = A(16x625 patches) x B(625x16 selected filters) via
// WMMA; out is the diagonal of D (pixel m's patch against pixel m's filter).
__global__ __launch_bounds__(256)
void gabor_route_wmma(const float* __restrict__ imgs,
                      const float* __restrict__ oris,
                      const float* __restrict__ kern,
                      float* __restrict__ out)
{
    __shared__ float tile[LH * LW];        // 5120 B image halo tile
    __shared__ float dbuf[TILE_H * 256];   // 8192 B: one 16x16 f32 D tile per wave

    const int tid = threadIdx.x;
    const int wv  = tid >> 5;              // wave id 0..7 -> pixel row within tile
    const int l   = tid & 31;              // lane id
    const int b   = blockIdx.z;
    const int x0  = blockIdx.x * TILE_W;
    const int y0  = blockIdx.y * TILE_H;

    // ---- stage image tile (zero-padded halo) into LDS ----
    const float* img = imgs + b * (IMG_H * IMG_W);
    for (int i = tid; i < LH * LW; i += 256) {
        int r  = i / LW, c = i - r * LW;
        int gy = y0 - HALO + r;
        int gx = x0 - HALO + c;
        float v = 0.f;
        if (gy >= 0 && gy < IMG_H && gx >= 0 && gx < IMG_W) v = img[gy * IMG_W + gx];
        tile[i] = v;
    }
    __syncthreads();

    // ---- per-lane orientation routing for pixel (x0 + (l&15), y0 + wv) ----
    // Lanes l and l+16 serve the same matrix row/column m = l&15, so both
    // compute the same selection (no cross-lane shuffle needed).
    const int m  = l & 15;
    const int px = x0 + m;
    const int py = y0 + wv;

    // bilinear upsample 32->256 (half-pixel centers, edge clamp == align_corners=False)
    float sx = (px + 0.5f) * 0.125f - 0.5f;
    float sy = (py + 0.5f) * 0.125f - 0.5f;
    float fxf = floorf(sx), fyf = floorf(sy);
    float fx = sx - fxf,    fy = sy - fyf;
    int ix0 = (int)fxf, iy0 = (int)fyf;
    int ix1 = ix0 + 1,  iy1 = iy0 + 1;
    ix0 = min(max(ix0, 0), ORI_W - 1); ix1 = min(max(ix1, 0), ORI_W - 1);
    iy0 = min(max(iy0, 0), ORI_H - 1); iy1 = min(max(iy1, 0), ORI_H - 1);
    const float* o0 = oris + (b * 2 + 0) * (ORI_H * ORI_W);
    const float* o1 = oris + (b * 2 + 1) * (ORI_H * ORI_W);
    float w00 = (1.f - fx) * (1.f - fy), w10 = fx * (1.f - fy);
    float w01 = (1.f - fx) * fy,         w11 = fx * fy;
    float v0 = o0[iy0*ORI_W+ix0]*w00 + o0[iy0*ORI_W+ix1]*w10
             + o0[iy1*ORI_W+ix0]*w01 + o0[iy1*ORI_W+ix1]*w11;
    float v1 = o1[iy0*ORI_W+ix0]*w00 + o1[iy0*ORI_W+ix1]*w10
             + o1[iy1*ORI_W+ix0]*w01 + o1[iy1*ORI_W+ix1]*w11;
    float ang = atan2f(v0, v1) * (90.0f / 3.14159265358979323846f) + 90.0f;
    int sel = ((int)ang % NANG + NANG) % NANG;     // guard fp edge at exactly -pi
    const float* kf = kern + sel * KK;
    __builtin_prefetch(kf, 0, 1);                  // global_prefetch_b8 (L2-resident bank)

    // ---- K reduction: 25 kernel rows, two accumulators for WMMA ILP ----
    const int  kcol = (l >> 4) << 1;               // 0 (lanes 0-15) / 2 (lanes 16-31)
    const bool lo   = (l < 16);
    const float* ta = tile + wv * LW + m;

    v8f acc0 = {}, acc1 = {};
    for (int ky = 0; ky < 24; ky += 2) {
        acc0 = gabor_row(ta + ky * LW,       kf + ky * KS,       kcol, lo, acc0);
        acc1 = gabor_row(ta + (ky + 1) * LW, kf + (ky + 1) * KS, kcol, lo, acc1);
    }
    acc0 = gabor_row(ta + 24 * LW, kf + 24 * KS, kcol, lo, acc0);
    v8f acc = acc0 + acc1;

    // ---- extract diagonal of D via LDS ----
    // C/D layout: VGPR v, lanes 0-15 -> (M=v, N=lane); lanes 16-31 -> (M=v+8, N=lane-16).
    float* dw = dbuf + wv * 256;
    #pragma unroll
    for (int v = 0; v < 8; ++v) dw[v * 32 + l] = acc[v];
    __syncthreads();
    if (l < 16) {
        int vv = (l < 8) ? l : (l - 8);
        int sl = (l < 8) ? l : (l + 16);
        out[b * (IMG_H * IMG_W) + py * IMG_W + px] = dw[vv * 32 + sl];
    }
}

extern "C" void kernel_launch(void* const* d_in, const int* in_sizes, int n_in,
                              void* d_out, int out_size, void* d_ws, size_t ws_size,
                              hipStream_t stream) {
    const float* imgs = (const float*)d_in[0];   // [B,1,256,256] f32
    const float* oris = (const float*)d_in[1];   // [B,2,32,32]   f32
    const float* kern = (const float*)d_in[2];   // [180,1,25,25] f32
    float* out = (float*)d_out;                  // [B,1,256,256] f32
    int B = in_sizes[0] / (IMG_H * IMG_W);
    dim3 grid(IMG_W / TILE_W, IMG_H / TILE_H, B);
    gabor_route_wmma<<<grid, 256, 0, stream>>>(imgs, oris, kern, out);
    (void)n_in; (void)out_size; (void)d_ws; (void)ws_size;
}